// BinaryNADE_87866440942052
// MI455X (gfx1250) — compile-verified
//
#include <hip/hip_runtime.h>
#include <cstdint>

// BinaryNADE forward on gfx1250 (MI455X).
// Bandwidth-bound: ~288MB mandatory fp32 traffic -> ~12.4us floor @ 23.3TB/s.
// Pipeline:
//   K1: base = c + W_ctx @ context        (V_WMMA_F32_16X16X4_F32, col-0 matvec)
//   K2: segment carries of the exclusive scan (wave32 shfl scan)
//   K3: main sweep: sigmoid + U contraction   (TDM tensor_load_to_lds double-buffer)
//   K4: p_val product reduction

typedef float v2f __attribute__((ext_vector_type(2)));
typedef float v8f __attribute__((ext_vector_type(8)));
typedef unsigned int v4u __attribute__((ext_vector_type(4)));
typedef int v4i __attribute__((ext_vector_type(4)));
typedef int v8i __attribute__((ext_vector_type(8)));

#define DIM_SOL 8192
#define DIM_CTX 2048
#define DIM_HID 4096
#define W_COLS (DIM_CTX + DIM_SOL)   // 10240
#define SEG 32
#define NSEG (DIM_SOL / SEG)         // 256

// ---------------------------------------------------------------------------
// Kernel 1: base[h] = c[h] + dot(W[h, 0:2048], context)  via WMMA f32 16x16x4.
// One wave per 16-row tile. B column N=0 carries context; other cols zero.
// Layouts (ISA 7.12.2): A 16x4: lanes0-15 comp{0,1}->K{0,1}, lanes16-31 ->K{2,3}.
// B 4x16: row K striped across lanes, K = comp + 2*(lane>=16), N = lane&15.
// D: lanes0-15 N=lane, VGPR r -> M=r; lanes16-31 -> M=r+8.
// ctx is loaded unconditionally (L1-resident) and masked with a select so the
// loop stays branch-free (v_cndmask, not s_and_saveexec).
// ---------------------------------------------------------------------------
__global__ __launch_bounds__(32) void base_wmma_kernel(
    const float* __restrict__ W, const float* __restrict__ ctx,
    const float* __restrict__ cvec, float* __restrict__ base)
{
    const int lane = threadIdx.x;
    const int m0   = blockIdx.x * 16;
    const int half = lane >> 4;          // 0 or 1
    const int koff = half * 2;           // K of vector component .x
    const int mrow = m0 + (lane & 15);
    const float* wrow = W + (size_t)mrow * W_COLS;
    const bool isN0 = ((lane & 15) == 0);

    v8f acc = {0.f, 0.f, 0.f, 0.f, 0.f, 0.f, 0.f, 0.f};
    for (int k = 0; k < DIM_CTX; k += 4) {
        const float2 av = *(const float2*)(wrow + k + koff);
        const float2 cc = *(const float2*)(ctx + k + koff);
        v2f a;
        a.x = av.x;
        a.y = av.y;
        v2f b;
        b.x = isN0 ? cc.x : 0.f;         // register select, no EXEC branching
        b.y = isN0 ? cc.y : 0.f;
        acc = __builtin_amdgcn_wmma_f32_16x16x4_f32(
            false, a, false, b, (short)0, acc, false, false);
    }
    if (isN0) {
#pragma unroll
        for (int r = 0; r < 8; ++r) {
            const int row = m0 + half * 8 + r;
            base[row] = acc[r] + cvec[row];
        }
    }
}

// ---------------------------------------------------------------------------
// Kernel 2: carry[h,s] = sum_{j < s*32} W_sol[h,j]*sol[j].
// One wave32 per row h; lane handles one 32-elem segment (contiguous 128B),
// wave shfl_up scan produces exclusive carries, serial across 8 groups.
// ---------------------------------------------------------------------------
__global__ __launch_bounds__(256) void carry_kernel(
    const float* __restrict__ W, const float* __restrict__ sol,
    float* __restrict__ carry)
{
    __shared__ float s_sol[DIM_SOL];
    const int t = threadIdx.x;
    const float4* solv = (const float4*)sol;
#pragma unroll
    for (int i = 0; i < 8; ++i)
        ((float4*)s_sol)[t + i * 256] = solv[t + i * 256];
    __syncthreads();

    const int lane = t & 31;
    const int wid  = t >> 5;
    const int h    = blockIdx.x * 8 + wid;
    const float4* rowv = (const float4*)(W + (size_t)h * W_COLS + DIM_CTX);

    float running = 0.f;
    for (int g = 0; g < 8; ++g) {
        const int s  = g * 32 + lane;
        const int e0 = s * SEG;
        float partial = 0.f;
#pragma unroll
        for (int i = 0; i < 8; ++i) {
            const float4 w = rowv[e0 / 4 + i];
            partial += w.x * s_sol[e0 + i * 4 + 0];
            partial += w.y * s_sol[e0 + i * 4 + 1];
            partial += w.z * s_sol[e0 + i * 4 + 2];
            partial += w.w * s_sol[e0 + i * 4 + 3];
        }
        float inc = partial;
#pragma unroll
        for (int off = 1; off < 32; off <<= 1) {
            const float v = __shfl_up(inc, off);
            if (lane >= off) inc += v;
        }
        carry[(size_t)h * NSEG + s] = running + (inc - partial);
        running += __shfl(inc, 31);
    }
}

// ---------------------------------------------------------------------------
// TDM helper: DMA a rows x rowlen fp32 tile (row stride stride_elems) into LDS.
// D# built per ISA ch.8: group0 {count, lds_addr, global_addr, type=2},
// group1 {data_size=4B, tensor_dim0/1, tile_dim0/1, tensor_dim0_stride}.
// This toolchain's builtin takes 6 args (g0 v4u, g1 v8i, g2 v4i, g3 v4i,
// extra v8i, cpol i32) -- extra group words zero for a plain 2D tile.
// ---------------------------------------------------------------------------
__device__ __forceinline__ void tdm_load_tile(unsigned lds_off,
                                              const float* gptr,
                                              unsigned rows, unsigned rowlen,
                                              unsigned stride_elems)
{
    const unsigned long long ga = (unsigned long long)(uintptr_t)gptr;
    v4u g0;
    g0.x = 1u;                                         // count=1, gather off
    g0.y = lds_off;                                    // lds_addr (bytes)
    g0.z = (unsigned)(ga & 0xffffffffull);             // global_addr[31:0]
    g0.w = (unsigned)((ga >> 32) & 0x1ffffffull)       // global_addr[56:32]
         | 0x80000000u;                                // type=2 ("image")
    v8i g1;
    g1[0] = (int)(2u << 16);                           // data_size = 4 bytes
    g1[1] = (int)((rowlen & 0xffffu) << 16);           // tensor_dim0[15:0]
    g1[2] = (int)(((rowlen >> 16) & 0xffffu)           // tensor_dim0[31:16]
                | ((rows & 0xffffu) << 16));           // tensor_dim1[15:0]
    g1[3] = (int)(((rows >> 16) & 0xffffu)             // tensor_dim1[31:16]
                | ((rowlen & 0xffffu) << 16));         // tile_dim0
    g1[4] = (int)(rows & 0xffffu);                     // tile_dim1 (tile_dim2=0)
    g1[5] = (int)stride_elems;                         // tensor_dim0_stride[31:0]
    g1[6] = 0;                                         // stride hi / dim1_stride lo
    g1[7] = 0;
    const v4i gz4 = {0, 0, 0, 0};
    const v8i gz8 = {0, 0, 0, 0, 0, 0, 0, 0};
    __builtin_amdgcn_tensor_load_to_lds(g0, g1, gz4, gz4, gz8, 0);
}

// ---------------------------------------------------------------------------
// Kernel 3: block = one 32-wide d-segment; sweep all 4096 h rows in 16 tiles.
// U tiles (32 x 256, stride 4096) streamed via TDM, double-buffered in LDS.
// W_sol rows read as per-thread B128 (full cachelines). Deterministic
// shuffle + LDS tree reduction -> p_dist into out[1..8192].
// ---------------------------------------------------------------------------
__global__ __launch_bounds__(256) void main_kernel(
    const float* __restrict__ W, const float* __restrict__ U,
    const float* __restrict__ sol, const float* __restrict__ bvec,
    const float* __restrict__ base, const float* __restrict__ carry,
    float* __restrict__ out)
{
    __shared__ float s_U[2][SEG * 256];   // 2 x 32KB double buffer
    __shared__ float s_sol[SEG];
    __shared__ float s_red[8][SEG];

    const int t    = threadIdx.x;
    const int seg  = blockIdx.x;
    const int d0   = seg * SEG;
    const int lane = t & 31;
    const int wid  = t >> 5;

    if (t < SEG) s_sol[t] = sol[d0 + t];

    if (wid == 0) {
        tdm_load_tile((unsigned)(uintptr_t)&s_U[0][0],
                      U + (size_t)d0 * DIM_HID, SEG, 256, DIM_HID);
    }

    float acc[SEG];
#pragma unroll
    for (int i = 0; i < SEG; ++i) acc[i] = 0.f;

    for (int ht = 0; ht < 16; ++ht) {
        if (wid == 0) {
            if (ht + 1 < 16) {
                tdm_load_tile((unsigned)(uintptr_t)&s_U[(ht + 1) & 1][0],
                              U + (size_t)d0 * DIM_HID + (ht + 1) * 256,
                              SEG, 256, DIM_HID);
                __builtin_amdgcn_s_wait_tensorcnt(1);  // current tile done
            } else {
                __builtin_amdgcn_s_wait_tensorcnt(0);
            }
        }
        __syncthreads();

        const float* su = &s_U[ht & 1][0];
        const int h = ht * 256 + t;
        const float4* rwv =
            (const float4*)(W + (size_t)h * W_COLS + DIM_CTX + d0);
        float4 wv[8];
#pragma unroll
        for (int i = 0; i < 8; ++i) wv[i] = rwv[i];

        float run = base[h] + carry[(size_t)h * NSEG + seg];
#pragma unroll
        for (int i = 0; i < SEG; ++i) {
            const float w  = ((const float*)wv)[i];
            const float Hs = 1.0f / (1.0f + __expf(-run));  // sigmoid(A[h,d0+i])
            acc[i] += su[i * 256 + t] * Hs;                 // U[d0+i,h]*H
            run += w * s_sol[i];                            // exclusive update
        }
        __syncthreads();   // buffer (ht&1) free before its next TDM overwrite
    }

    // deterministic reduction over 256 threads (sum over h)
#pragma unroll
    for (int i = 0; i < SEG; ++i) {
        float v = acc[i];
#pragma unroll
        for (int off = 16; off > 0; off >>= 1) v += __shfl_down(v, off);
        if (lane == 0) s_red[wid][i] = v;
    }
    __syncthreads();
    if (t < SEG) {
        float tot = 0.f;
#pragma unroll
        for (int w8 = 0; w8 < 8; ++w8) tot += s_red[w8][t];
        const float p = 1.0f / (1.0f + __expf(-(bvec[d0 + t] + tot)));
        out[1 + d0 + t] = p;
    }
}

// ---------------------------------------------------------------------------
// Kernel 4: p_val = prod_d [ p_d^sol_d + (1-p_0)^(1-sol_d) ]  (ref semantics,
// including its use of p_dist[0] in the second term). Deterministic tree.
// ---------------------------------------------------------------------------
__global__ __launch_bounds__(256) void finalize_kernel(
    const float* __restrict__ sol, float* __restrict__ out)
{
    __shared__ float s_m[256];
    const int t = threadIdx.x;
    const float p0 = out[1];
    float partial = 1.0f;
    for (int i = 0; i < 32; ++i) {
        const int d = t * 32 + i;
        const float s = sol[d];
        const float p = out[1 + d];
        partial *= powf(p, s) + powf(1.0f - p0, 1.0f - s);
    }
    s_m[t] = partial;
    __syncthreads();
    for (int off = 128; off > 0; off >>= 1) {
        if (t < off) s_m[t] *= s_m[t + off];
        __syncthreads();
    }
    if (t == 0) out[0] = s_m[0];
}

extern "C" void kernel_launch(void* const* d_in, const int* in_sizes, int n_in,
                              void* d_out, int out_size, void* d_ws, size_t ws_size,
                              hipStream_t stream)
{
    (void)in_sizes; (void)n_in; (void)out_size; (void)ws_size;
    const float* ctx = (const float*)d_in[0];   // (2048,)
    const float* sol = (const float*)d_in[1];   // (8192,)
    const float* W   = (const float*)d_in[2];   // (4096, 10240)
    const float* U   = (const float*)d_in[3];   // (8192, 4096)
    const float* b   = (const float*)d_in[4];   // (8192,)
    const float* c   = (const float*)d_in[5];   // (4096,)
    float* out = (float*)d_out;                 // [p_val, p_dist(8192)]

    float* base  = (float*)d_ws;                // 4096 floats
    float* carry = (float*)d_ws + DIM_HID;      // 4096*256 floats (~4MB)

    base_wmma_kernel<<<DIM_HID / 16, 32, 0, stream>>>(W, ctx, c, base);
    carry_kernel<<<DIM_HID / 8, 256, 0, stream>>>(W, sol, carry);
    main_kernel<<<NSEG, 256, 0, stream>>>(W, U, sol, b, base, carry, out);
    finalize_kernel<<<1, 256, 0, stream>>>(sol, out);
}